// GFCNC_17867063951704
// MI455X (gfx1250) — compile-verified
//
#include <hip/hip_runtime.h>
#include <math.h>

typedef float v2f __attribute__((ext_vector_type(2)));
typedef float v8f __attribute__((ext_vector_type(8)));

__device__ __forceinline__ float eluf(float x) { return x > 0.f ? x : expm1f(x); }

// ---------------------------------------------------------------- utilities
__global__ void fill_kernel(float* __restrict__ p, float v, int n) {
    int i = blockIdx.x * blockDim.x + threadIdx.x;
    if (i < n) p[i] = v;
}

__global__ void deg_kernel(const int* __restrict__ ei, int E, float* __restrict__ deg) {
    int e = blockIdx.x * blockDim.x + threadIdx.x;
    if (e < E) atomicAdd(&deg[ei[E + e]], 1.0f);
}

__device__ __forceinline__ void atomicMaxF(float* addr, float val) {
    unsigned int* ua = (unsigned int*)addr;
    unsigned int old = *ua;
    while (__uint_as_float(old) < val) {
        unsigned int assumed = old;
        old = atomicCAS(ua, assumed, __float_as_uint(val));
        if (old == assumed) break;
    }
}

__global__ void pool_max_kernel(const float* __restrict__ X, const int* __restrict__ cl,
                                float* __restrict__ out, int N, int C) {
    long long tid = (long long)blockIdx.x * blockDim.x + threadIdx.x;
    if (tid >= (long long)N * C) return;
    int n = (int)(tid / C), c = (int)(tid % C);
    atomicMaxF(&out[(long long)cl[n] * C + c], X[tid]);
}

__global__ void gather_add_kernel(const float* __restrict__ coarse, const int* __restrict__ cl,
                                  const float* __restrict__ skip, float* __restrict__ out,
                                  int N, int C) {
    long long tid = (long long)blockIdx.x * blockDim.x + threadIdx.x;
    if (tid >= (long long)N * C) return;
    int n = (int)(tid / C), c = (int)(tid % C);
    float v = coarse[(long long)cl[n] * C + c];
    if (skip) v += skip[tid];
    out[tid] = v;
}

// ------------------------------------------------- BatchNorm (training mode)
__global__ void bn_kernel(float* __restrict__ X, const float* __restrict__ gamma,
                          const float* __restrict__ beta, int N, int C) {
    int c = blockIdx.x;
    int t = threadIdx.x;
    __shared__ float sdata[256];
    float s = 0.f;
    for (int n = t; n < N; n += 256) s += X[(long long)n * C + c];
    sdata[t] = s;
    __syncthreads();
    for (int o = 128; o > 0; o >>= 1) { if (t < o) sdata[t] += sdata[t + o]; __syncthreads(); }
    float mu = sdata[0] / (float)N;
    __syncthreads();
    float s2 = 0.f;
    for (int n = t; n < N; n += 256) { float d = X[(long long)n * C + c] - mu; s2 += d * d; }
    sdata[t] = s2;
    __syncthreads();
    for (int o = 128; o > 0; o >>= 1) { if (t < o) sdata[t] += sdata[t + o]; __syncthreads(); }
    float var = sdata[0] / (float)N;
    float sc = rsqrtf(var + 1e-5f) * gamma[c];
    float bo = beta[c];
    for (int n = t; n < N; n += 256) {
        long long idx = (long long)n * C + c;
        X[idx] = (X[idx] - mu) * sc + bo;
    }
}

// ------------------------------------- SplineConv messages, K>1 (scalar path)
__global__ void msg_scalar(const float* __restrict__ X, const int* __restrict__ ei,
                           const float* __restrict__ ea, const float* __restrict__ W,
                           float* __restrict__ agg, int E, int Cin, int Cout, int K) {
    long long tid = (long long)blockIdx.x * blockDim.x + threadIdx.x;
    if (tid >= (long long)E * Cout) return;
    int e = (int)(tid / Cout);
    int co = (int)(tid % Cout);
    int src = ei[e], dst = ei[E + e];
    float Km1 = (float)(K - 1);
    float v0 = ea[2 * e] * Km1, v1 = ea[2 * e + 1] * Km1;
    float f0 = fminf(fmaxf(floorf(v0), 0.f), (float)(K - 2));
    float f1 = fminf(fmaxf(floorf(v1), 0.f), (float)(K - 2));
    float r0 = v0 - f0, r1 = v1 - f1;
    int i0 = (int)f0, i1 = (int)f1;
    const float* xr = X + (long long)src * Cin;
    float acc = 0.f;
#pragma unroll
    for (int i = 0; i < 2; i++) {
        float w0 = i ? r0 : 1.f - r0;
#pragma unroll
        for (int j = 0; j < 2; j++) {
            float w1 = j ? r1 : 1.f - r1;
            float w = w0 * w1;
            if (w != 0.f) {
                int kidx = (i0 + i) + (i1 + j) * K;
                const float* Wk = W + (long long)kidx * Cin * Cout + co;
                float s = 0.f;
                for (int c = 0; c < Cin; c++) s += xr[c] * Wk[(long long)c * Cout];
                acc += w * s;
            }
        }
    }
    atomicAdd(&agg[(long long)dst * Cout + co], acc);
}

// ------------------------------ SplineConv messages, K==1: WMMA gather-GEMM
// One wave per block; 16 edges x 16 couts per tile; fp32 WMMA 16x16x4.
// Requires Cin % 4 == 0. Branchless loop body: OOB rows/cols are clamped to
// valid indices (their garbage results are discarded by the guarded store).
__global__ void msg_wmma_k1(const float* __restrict__ X, const int* __restrict__ ei,
                            const float* __restrict__ W0, float* __restrict__ agg,
                            int E, int Cin, int Cout) {
    int lane = threadIdx.x;          // 0..31
    int m = lane & 15, half = lane >> 4;
    int e0 = blockIdx.x * 16;
    int col0 = blockIdx.y * 16;
    int eA = min(e0 + m, E - 1);
    int srow = ei[eA];
    int col = col0 + m;
    int colc = min(col, Cout - 1);
    const float* pA = X + (long long)srow * Cin + 2 * half;          // even offset: b64 ok
    const float* pB = W0 + (long long)(2 * half) * Cout + colc;
    long long stepB = 4LL * Cout;
    v8f acc = {};
#pragma unroll 4
    for (int kb = 0; kb < Cin; kb += 4) {
        v2f a = *(const v2f*)(pA + kb);
        v2f b;
        b.x = pB[0];
        b.y = pB[Cout];
        pB += stepB;
        acc = __builtin_amdgcn_wmma_f32_16x16x4_f32(false, a, false, b, (short)0, acc,
                                                    false, false);
    }
    bool vc = col < Cout;
#pragma unroll
    for (int r = 0; r < 8; r++) {
        int er = e0 + r + 8 * half;
        if (er < E && vc) {
            int dst = ei[E + er];
            atomicAdd(&agg[(long long)dst * Cout + col], acc[r]);
        }
    }
}

// --------- finalize: out = agg/max(deg,1) + X@root + bias, optional ELU (WMMA)
// Requires Cin % 4 == 0 (host dispatches finalize_scalar otherwise).
__global__ void finalize_wmma(const float* __restrict__ X, const float* __restrict__ root,
                              const float* __restrict__ bias, const float* __restrict__ agg,
                              const float* __restrict__ deg, float* __restrict__ out,
                              int N, int Cin, int Cout, int applyElu) {
    int lane = threadIdx.x;
    int m = lane & 15, half = lane >> 4;
    int row0 = blockIdx.x * 16;
    int col0 = blockIdx.y * 16;
    int rowA = min(row0 + m, N - 1);
    int col = col0 + m;
    int colc = min(col, Cout - 1);
    const float* pA = X + (long long)rowA * Cin + 2 * half;
    const float* pB = root + (long long)(2 * half) * Cout + colc;
    long long stepB = 4LL * Cout;
    v8f acc = {};
#pragma unroll 4
    for (int kb = 0; kb < Cin; kb += 4) {
        v2f a = *(const v2f*)(pA + kb);
        v2f b;
        b.x = pB[0];
        b.y = pB[Cout];
        pB += stepB;
        acc = __builtin_amdgcn_wmma_f32_16x16x4_f32(false, a, false, b, (short)0, acc,
                                                    false, false);
    }
    bool vc = col < Cout;
    float bv = vc ? bias[colc] : 0.f;
#pragma unroll
    for (int r = 0; r < 8; r++) {
        int row = row0 + r + 8 * half;
        if (row < N && vc) {
            float d = deg[row];
            float val = agg[(long long)row * Cout + col] / fmaxf(d, 1.f) + acc[r] + bv;
            if (applyElu) val = eluf(val);
            out[(long long)row * Cout + col] = val;
        }
    }
}

// --------- finalize for tiny Cin (Cin % 4 != 0, i.e. conv1a's Cin==1): scalar
__global__ void finalize_scalar(const float* __restrict__ X, const float* __restrict__ root,
                                const float* __restrict__ bias, const float* __restrict__ agg,
                                const float* __restrict__ deg, float* __restrict__ out,
                                int N, int Cin, int Cout, int applyElu) {
    long long tid = (long long)blockIdx.x * blockDim.x + threadIdx.x;
    if (tid >= (long long)N * Cout) return;
    int row = (int)(tid / Cout), col = (int)(tid % Cout);
    float s = 0.f;
    for (int c = 0; c < Cin; c++) s += X[(long long)row * Cin + c] * root[(long long)c * Cout + col];
    float val = agg[tid] / fmaxf(deg[row], 1.f) + s + bias[col];
    if (applyElu) val = eluf(val);
    out[tid] = val;
}

// =========================================================== host orchestration
static inline int ceil_div_ll(long long a, long long b) { return (int)((a + b - 1) / b); }

extern "C" void kernel_launch(void* const* d_in, const int* in_sizes, int n_in,
                              void* d_out, int out_size, void* d_ws, size_t ws_size,
                              hipStream_t stream) {
    const int* ei[5]; const float* ea[5]; const int* cl[4];
    const float *W[12], *R[12], *B[12], *G[4], *Bt[4]; const float* x;
    int E[5];
    // conv order: 0:conv1a 1:conv1b 2:conv2a 3:conv2b 4:conv3a 5:conv3b
    //             6:conv4a 7:conv4b 8:score_fr 9:score_pool3 10:score_pool2 11:convout
    bool insertion = (in_sizes[14] == 800);   // conv1a.W (25*1*32) vs bn1.gamma (32)
    if (insertion) {
        for (int l = 0; l < 5; l++) { ei[l] = (const int*)d_in[2 * l]; ea[l] = (const float*)d_in[2 * l + 1]; }
        for (int l = 0; l < 4; l++) cl[l] = (const int*)d_in[10 + l];
        for (int i = 0; i < 12; i++) {
            W[i] = (const float*)d_in[14 + 3 * i];
            R[i] = (const float*)d_in[15 + 3 * i];
            B[i] = (const float*)d_in[16 + 3 * i];
        }
        for (int i = 0; i < 4; i++) { G[i] = (const float*)d_in[50 + 2 * i]; Bt[i] = (const float*)d_in[51 + 2 * i]; }
        x = (const float*)d_in[58];
        for (int l = 0; l < 5; l++) E[l] = in_sizes[2 * l] / 2;
    } else {  // sorted-key pytree flattening
        for (int l = 0; l < 4; l++) cl[l] = (const int*)d_in[l];
        for (int l = 0; l < 5; l++) { ea[l] = (const float*)d_in[4 + l]; ei[l] = (const int*)d_in[9 + l]; }
        for (int i = 0; i < 4; i++) { G[i] = (const float*)d_in[14 + 2 * i]; Bt[i] = (const float*)d_in[15 + 2 * i]; }
        const int slot[12] = {22, 25, 28, 31, 34, 37, 40, 43, 49, 55, 52, 46};
        for (int i = 0; i < 12; i++) {
            W[i] = (const float*)d_in[slot[i]];
            R[i] = (const float*)d_in[slot[i] + 1];
            B[i] = (const float*)d_in[slot[i] + 2];
        }
        x = (const float*)d_in[58];
        for (int l = 0; l < 5; l++) E[l] = in_sizes[9 + l] / 2;
    }
    const int N[5] = {36864, 9216, 2304, 576, 144};

    // ---- workspace arena
    float* ws = (float*)d_ws;
    size_t off = 0;
    auto alloc = [&](size_t n) { float* p = ws + off; off += n; return p; };
    float* agg = alloc(36864u * 32);
    float* h0a = alloc(36864u * 32);
    float* h0b = alloc(36864u * 32);
    float* h1  = alloc(9216u * 32);
    float* h1a = alloc(9216u * 64);
    float* h1b = alloc(9216u * 64);
    float* h2  = alloc(2304u * 64);    // pool2 (persists)
    float* h2a = alloc(2304u * 128);
    float* h2b = alloc(2304u * 128);
    float* h3  = alloc(576u * 128);    // pool3 (persists)
    float* h3a = alloc(576u * 256);
    float* h3b = alloc(576u * 256);
    float* h4  = alloc(144u * 256);
    float* sc  = alloc(144u * 32);
    float* u3  = alloc(576u * 32);
    float* u2  = alloc(2304u * 32);
    float* deg0 = alloc(36864); float* deg1 = alloc(9216); float* deg2 = alloc(2304);
    float* deg3 = alloc(576);   float* deg4 = alloc(144);
    float* u1 = h1a;  // dead after conv2b
    float* u0 = h0a;  // dead after conv1b
    float* degs[5] = {deg0, deg1, deg2, deg3, deg4};

    const int TB = 256;
    auto filln = [&](float* p, long long n, float v) {
        fill_kernel<<<ceil_div_ll(n, TB), TB, 0, stream>>>(p, v, (int)n);
    };

    for (int l = 0; l < 5; l++) {
        filln(degs[l], N[l], 0.f);
        deg_kernel<<<ceil_div_ll(E[l], TB), TB, 0, stream>>>(ei[l], E[l], degs[l]);
    }

    auto conv = [&](int ci, int lvl, const float* X, float* Y, int Cin, int Cout, int K, bool elu) {
        filln(agg, (long long)N[lvl] * Cout, 0.f);
        if (K == 1) {
            dim3 g((E[lvl] + 15) / 16, (Cout + 15) / 16);
            msg_wmma_k1<<<g, 32, 0, stream>>>(X, ei[lvl], W[ci], agg, E[lvl], Cin, Cout);
        } else {
            long long tot = (long long)E[lvl] * Cout;
            msg_scalar<<<ceil_div_ll(tot, TB), TB, 0, stream>>>(X, ei[lvl], ea[lvl], W[ci], agg,
                                                                E[lvl], Cin, Cout, K);
        }
        if (Cin % 4 == 0) {
            dim3 gf((N[lvl] + 15) / 16, (Cout + 15) / 16);
            finalize_wmma<<<gf, 32, 0, stream>>>(X, R[ci], B[ci], agg, degs[lvl], Y,
                                                 N[lvl], Cin, Cout, elu ? 1 : 0);
        } else {
            long long tot = (long long)N[lvl] * Cout;
            finalize_scalar<<<ceil_div_ll(tot, TB), TB, 0, stream>>>(X, R[ci], B[ci], agg,
                                                                     degs[lvl], Y, N[lvl], Cin,
                                                                     Cout, elu ? 1 : 0);
        }
    };
    auto pool = [&](const float* Xp, int lvl, float* Y, int C) {
        filln(Y, (long long)N[lvl + 1] * C, -INFINITY);
        long long tot = (long long)N[lvl] * C;
        pool_max_kernel<<<ceil_div_ll(tot, TB), TB, 0, stream>>>(Xp, cl[lvl], Y, N[lvl], C);
    };

    // ---------------- encoder
    conv(0, 0, x,   h0a, 1,   32, 5, true);                      // conv1a (scalar finalize)
    conv(1, 0, h0a, h0b, 32,  32, 5, true);                      // conv1b
    bn_kernel<<<32, 256, 0, stream>>>(h0b, G[0], Bt[0], N[0], 32);
    pool(h0b, 0, h1, 32);
    conv(2, 1, h1,  h1a, 32,  64, 3, true);                      // conv2a
    conv(3, 1, h1a, h1b, 64,  64, 3, true);                      // conv2b
    bn_kernel<<<64, 256, 0, stream>>>(h1b, G[1], Bt[1], N[1], 64);
    pool(h1b, 1, h2, 64);                                        // pool2
    conv(4, 2, h2,  h2a, 64, 128, 3, true);                      // conv3a
    conv(5, 2, h2a, h2b, 128, 128, 1, true);                     // conv3b (WMMA msg)
    bn_kernel<<<128, 256, 0, stream>>>(h2b, G[2], Bt[2], N[2], 128);
    pool(h2b, 2, h3, 128);                                       // pool3
    conv(6, 3, h3,  h3a, 128, 256, 1, true);                     // conv4a (WMMA msg)
    conv(7, 3, h3a, h3b, 256, 256, 1, true);                     // conv4b (WMMA msg)
    bn_kernel<<<256, 256, 0, stream>>>(h3b, G[3], Bt[3], N[3], 256);
    pool(h3b, 3, h4, 256);
    conv(8, 4, h4, sc, 256, 32, 1, true);                        // score_fr (WMMA msg)

    // ---------------- decoder: unpool + skips
    conv(9, 3, h3, u3, 128, 32, 3, true);                        // score_pool3(pool3)
    gather_add_kernel<<<ceil_div_ll(576LL * 32, TB), TB, 0, stream>>>(sc, cl[3], u3, u3, 576, 32);
    conv(10, 2, h2, u2, 64, 32, 3, true);                        // score_pool2(pool2)
    gather_add_kernel<<<ceil_div_ll(2304LL * 32, TB), TB, 0, stream>>>(u3, cl[2], u2, u2, 2304, 32);
    gather_add_kernel<<<ceil_div_ll(9216LL * 32, TB), TB, 0, stream>>>(u2, cl[1], nullptr, u1, 9216, 32);
    gather_add_kernel<<<ceil_div_ll(36864LL * 32, TB), TB, 0, stream>>>(u1, cl[0], nullptr, u0, 36864, 32);

    conv(11, 0, u0, (float*)d_out, 32, 1, 5, false);             // convout -> [36864,1]
}